// RTDETRPostProcessor_48627619726099
// MI455X (gfx1250) — compile-verified
//
#include <hip/hip_runtime.h>
#include <hip/hip_bf16.h>
#include <stdint.h>

// ---- Problem constants (match reference) ----
#define NUM_CLASSES 80
#define NQ          2000
#define NB          128
#define ROW         (NQ * NUM_CLASSES)   // 160000 scores per batch
#define TOPK        300
#define NTHREADS    1024                 // 32 wave32 waves per workgroup
#define KEYBITS     13
#define NBUCKETS    (1 << KEYBITS)       // 8192 buckets (32 KB LDS)
#define KEYSHIFT    (32 - KEYBITS)
#define CHUNK       8192                 // floats per TDM chunk (32 KB)
#define NCHUNK      ((ROW + CHUNK - 1) / CHUNK)   // 20
#define CAND_CAP    2048

typedef __attribute__((ext_vector_type(4))) unsigned int u32x4;
typedef __attribute__((ext_vector_type(8))) int          i32x8;
typedef __attribute__((ext_vector_type(4))) int          i32x4;

__device__ __forceinline__ unsigned umin_u(unsigned a, unsigned b) { return a < b ? a : b; }

// Order-preserving map: float bits -> unsigned, so integer compare == float compare.
__device__ __forceinline__ unsigned monotone(unsigned u) {
    return (u & 0x80000000u) ? ~u : (u | 0x80000000u);
}

// Issue one TDM load: nelem fp32 from gptr -> LDS byte offset lds_off.
// D# built per CDNA5 ISA ch.8 (group0 128b, group1 256b; groups 2/3 zero => <=2D).
__device__ __forceinline__ void tdm_load(unsigned lds_off, const float* gptr, unsigned nelem) {
    unsigned long long ga = (unsigned long long)(uintptr_t)gptr;
    u32x4 g0;
    g0[0] = 1u;                                             // count=1, user mode
    g0[1] = lds_off;                                        // lds_addr [63:32]
    g0[2] = (unsigned)(ga & 0xFFFFFFFFu);                   // global_addr lo
    g0[3] = (unsigned)((ga >> 32) & 0x01FFFFFFu)            // global_addr[56:32]
          | (2u << 30);                                     // type=2 ("image")
    i32x8 g1;
    g1[0] = (int)(2u << 16);                                // data_size=4B; no mcast/pad
    g1[1] = (int)((nelem & 0xFFFFu) << 16);                 // tensor_dim0 lo16 @ [63:48]
    g1[2] = (int)(((nelem >> 16) & 0xFFFFu) | (1u << 16));  // dim0 hi16 | tensor_dim1=1
    g1[3] = (int)((nelem & 0xFFFFu) << 16);                 // tile_dim0 @ [127:112]
    g1[4] = 1;                                              // tile_dim1=1, tile_dim2=0
    g1[5] = (int)nelem;                                     // tensor_dim0_stride lo32
    g1[6] = 0;
    g1[7] = 0;
    i32x4 z4 = {0, 0, 0, 0};
    i32x8 z8 = {0, 0, 0, 0, 0, 0, 0, 0};
    // amdgpu-toolchain (clang-23) 6-arg form
    __builtin_amdgcn_tensor_load_to_lds(g0, g1, z4, z4, z8, 0);
}

__global__ __launch_bounds__(NTHREADS)
void rtdetr_post_kernel(const float* __restrict__ logits,
                        const float* __restrict__ boxes,
                        const int*   __restrict__ sizes,
                        float*       __restrict__ out)
{
    __shared__ unsigned hist[NBUCKETS];                        // 32 KB
    __shared__ __align__(16) float stage[2][CHUNK];            // 64 KB (TDM double buffer)
    __shared__ unsigned segsum[256];
    __shared__ unsigned s_thr, s_above, s_need, s_ctrA, s_ctrC;
    __shared__ unsigned long long entry[TOPK];
    __shared__ unsigned candK[CAND_CAP];
    __shared__ unsigned candI[CAND_CAP];

    const int tid = threadIdx.x;
    const int b   = blockIdx.x;
    const float* row = logits + (size_t)b * ROW;

    for (int i = tid; i < NBUCKETS; i += NTHREADS) hist[i] = 0u;
    if (tid == 0) { s_ctrA = 0u; s_ctrC = 0u; }
    __syncthreads();

    const bool wave0 = (tid < 32);
    // Low 32 bits of a generic LDS pointer == LDS byte offset (aperture in high bits).
    const unsigned stOff0 = (unsigned)(uintptr_t)&stage[0][0];
    const unsigned stOff1 = (unsigned)(uintptr_t)&stage[1][0];

    // ---- Pass 1: TDM-streamed histogram of monotone key top bits ----
    if (wave0) tdm_load(stOff0, row, umin_u(CHUNK, ROW));
    for (int ch = 0; ch < NCHUNK; ++ch) {
        if (wave0) {
            if (ch + 1 < NCHUNK) {
                unsigned base = (unsigned)(ch + 1) * CHUNK;
                tdm_load(((ch + 1) & 1) ? stOff1 : stOff0, row + base,
                         umin_u(CHUNK, ROW - base));
                __builtin_amdgcn_s_wait_tensorcnt(1);   // chunk ch landed
            } else {
                __builtin_amdgcn_s_wait_tensorcnt(0);
            }
        }
        __syncthreads();                                 // data visible to all waves
        const float* bufp = stage[ch & 1];
        const unsigned n = umin_u(CHUNK, ROW - (unsigned)ch * CHUNK);
        for (unsigned j = tid; j < n; j += NTHREADS) {
            unsigned k = monotone(__float_as_uint(bufp[j])) >> KEYSHIFT;
            atomicAdd(&hist[k], 1u);
        }
        __syncthreads();                                 // safe to overwrite buffer
    }

    // ---- Find threshold bucket (two-level suffix scan) ----
    if (tid < 256) {
        unsigned s = 0;
        for (int j = 0; j < NBUCKETS / 256; ++j) s += hist[tid * (NBUCKETS / 256) + j];
        segsum[tid] = s;
    }
    __syncthreads();
    if (tid == 0) {
        unsigned cum = 0; int seg = 255;
        for (; seg > 0; --seg) {
            if (cum + segsum[seg] >= TOPK) break;
            cum += segsum[seg];
        }
        int bkt = seg * (NBUCKETS / 256) + (NBUCKETS / 256) - 1;
        for (; bkt > seg * (NBUCKETS / 256); --bkt) {
            if (cum + hist[bkt] >= TOPK) break;
            cum += hist[bkt];
        }
        s_thr   = (unsigned)bkt;
        s_above = cum;              // strictly-above count (< TOPK)
        s_need  = TOPK - cum;       // take from threshold bucket
    }
    __syncthreads();
    const unsigned thr = s_thr, above = s_above, need = s_need;

    // ---- Pass 2: collect winners + tie candidates (re-read from 192MB L2) ----
    const float4* row4 = reinterpret_cast<const float4*>(row);
    for (unsigned v = tid; v < ROW / 4; v += NTHREADS) {
        float4 f = row4[v];
        const float vals[4] = { f.x, f.y, f.z, f.w };
        #pragma unroll
        for (int c = 0; c < 4; ++c) {
            unsigned key = monotone(__float_as_uint(vals[c]));
            unsigned bk  = key >> KEYSHIFT;
            if (bk > thr) {
                unsigned slot = atomicAdd(&s_ctrA, 1u);
                if (slot < TOPK)
                    entry[slot] = ((unsigned long long)key << 32) | (unsigned)(~(4u * v + (unsigned)c));
            } else if (bk == thr) {
                unsigned slot = atomicAdd(&s_ctrC, 1u);
                if (slot < CAND_CAP) { candK[slot] = key; candI[slot] = 4u * v + (unsigned)c; }
            }
        }
    }
    __syncthreads();

    // ---- Resolve ties by exact rank (value desc, index asc — matches lax.top_k) ----
    const unsigned nc = umin_u(s_ctrC, CAND_CAP);
    for (unsigned j = tid; j < nc; j += NTHREADS) {
        unsigned long long me = ((unsigned long long)candK[j] << 32) | (unsigned)(~candI[j]);
        unsigned rank = 0;
        for (unsigned t = 0; t < nc; ++t) {
            unsigned long long o = ((unsigned long long)candK[t] << 32) | (unsigned)(~candI[t]);
            rank += (o > me) ? 1u : 0u;
        }
        if (rank < need) entry[above + rank] = me;
    }
    __syncthreads();

    // ---- Rank-sort the 300 winners and emit outputs ----
    if (tid < TOPK) {
        unsigned long long me = entry[tid];
        unsigned rank = 0;
        for (int t = 0; t < TOPK; ++t) rank += (entry[t] > me) ? 1u : 0u;

        unsigned key = (unsigned)(me >> 32);
        unsigned idx = ~(unsigned)(me & 0xFFFFFFFFull);
        unsigned u   = (key & 0x80000000u) ? (key & 0x7FFFFFFFu) : ~key;  // invert monotone map
        float logit  = __uint_as_float(u);
        float score  = 1.0f / (1.0f + __expf(-logit));

        unsigned label = idx % NUM_CLASSES;
        unsigned q     = idx / NUM_CLASSES;

        const float* bx = boxes + ((size_t)b * NQ + q) * 4;
        float cx = bx[0], cy = bx[1], w = bx[2], h = bx[3];
        float img_h = (float)sizes[b * 2 + 0];
        float img_w = (float)sizes[b * 2 + 1];
        float x0 = fmaxf((cx - 0.5f * w) * img_w, 0.0f);
        float y0 = fmaxf((cy - 0.5f * h) * img_h, 0.0f);
        float bw = fmaxf(w * img_w, 1.0f);
        float bh = fmaxf(h * img_h, 1.0f);

        const size_t o = (size_t)b * TOPK + rank;
        out[o] = (float)label;                               // labels  [B,300]
        float* ob = out + (size_t)NB * TOPK + o * 4;         // boxes   [B,300,4]
        ob[0] = x0; ob[1] = y0; ob[2] = bw; ob[3] = bh;
        out[(size_t)NB * TOPK * 5 + o] = score;              // scores  [B,300]
    }
}

extern "C" void kernel_launch(void* const* d_in, const int* in_sizes, int n_in,
                              void* d_out, int out_size, void* d_ws, size_t ws_size,
                              hipStream_t stream) {
    (void)in_sizes; (void)n_in; (void)d_ws; (void)ws_size; (void)out_size;
    const float* logits = (const float*)d_in[0];   // pred_logits [128,2000,80] f32
    const float* boxes  = (const float*)d_in[1];   // pred_boxes  [128,2000,4]  f32
    const int*   sizes  = (const int*)d_in[2];     // orig_target_sizes [128,2] i32
    float* out = (float*)d_out;
    rtdetr_post_kernel<<<NB, NTHREADS, 0, stream>>>(logits, boxes, sizes, out);
}